// CoGNN_70592082477654
// MI455X (gfx1250) — compile-verified
//
#include <hip/hip_runtime.h>
#include <hip/hip_bf16.h>

// ---------------------------------------------------------------------------
// CoGNN forward for MI455X (gfx1250, wave32).
//  * straight-through Gumbel => forward value is the hard one-hot; the
//    positive per-node temperature / softplus / Wt_* are dead code.
//  * ew[e] = gate_in[u]*gate_out[v] in {0,1}.
//  * Aggregations run as CSR gathers (CSR sorted by edge id => deterministic
//    fp summation order, zero fp atomics) instead of atomic scatters.
//  * GEMMs on V_WMMA_F32_16X16X4_F32, fp32 end-to-end; 32x64 strip per wave,
//    all fragment loads of a k-step grouped ahead of the 8 WMMAs so the
//    compiler can clause loads and stagger s_wait_loadcnt.
// ---------------------------------------------------------------------------

typedef float v2f __attribute__((ext_vector_type(2)));
typedef float v8f __attribute__((ext_vector_type(8)));

#define D_MODEL 256
#define D_INP   128

__device__ __forceinline__ float wave_sum32(float v) {
#pragma unroll
    for (int o = 16; o > 0; o >>= 1) v += __shfl_xor(v, o, 32);
    return v;
}

// ---------------------------------------------------------------------------
// WMMA fp32 GEMM: C[M,Ncols] = act( A@B (+ A2@B2) + bias ).
// One wave computes a (16*MT) x (16*NT) tile; A fragments shared across NT
// column tiles, B fragments shared across MT row tiles.
// A frag (ISA 7.12.2, 32-bit A 16x4): lane<16 -> M=lane, K=k+{0,1};
// lane>=16 -> M=lane-16, K=k+{2,3}. B mirrors. C/D: reg r -> row0+r (+8 for
// upper lane half), col = col0 + lane%16.
// ---------------------------------------------------------------------------
template <int MT, int NT, bool DUAL, bool RELU>
__global__ void gemm_wmma_f32(const float* __restrict__ A, int lda,
                              const float* __restrict__ B,
                              const float* __restrict__ A2,
                              const float* __restrict__ B2,
                              const float* __restrict__ bias,
                              float* __restrict__ C, int ldc,
                              int M, int K, int Ncols) {
    const int wavesPerBlock = blockDim.x >> 5;
    const int waveId = threadIdx.x >> 5;
    const int lane   = threadIdx.x & 31;
    const int groupsN = Ncols / (16 * NT);
    const int groupsM = (M / 16 + MT - 1) / MT;
    const int tile = blockIdx.x * wavesPerBlock + waveId;
    if (tile >= groupsM * groupsN) return;         // wave-uniform exit
    const int tm = tile / groupsN;
    const int tg = tile - tm * groupsN;
    const int row0 = tm * 16 * MT, col0 = tg * 16 * NT;
    const int half = lane >> 4;
    const int l16  = lane & 15;

    v8f acc[MT][NT];
#pragma unroll
    for (int m = 0; m < MT; ++m)
#pragma unroll
        for (int t = 0; t < NT; ++t) acc[m][t] = v8f{};

#pragma unroll
    for (int pass = 0; pass < (DUAL ? 2 : 1); ++pass) {
        const float* Asrc = pass ? A2 : A;
        const float* Bsrc = pass ? B2 : B;
        const int    alda = pass ? K : lda;
        const float* Ap[MT];
#pragma unroll
        for (int m = 0; m < MT; ++m) {
            int r = row0 + m * 16;
            if (r + 16 > M) r = M - 16;            // tail clamp (stores skipped)
            Ap[m] = Asrc + (size_t)(r + l16) * alda + 2 * half;
        }
        const float* Bp0 = Bsrc + (size_t)(2 * half) * Ncols + col0 + l16;
        for (int k = 0; k < K; k += 4) {
            // ---- group ALL fragment loads of this k-step ----
            v2f a[MT];
#pragma unroll
            for (int m = 0; m < MT; ++m) { a[m].x = Ap[m][k]; a[m].y = Ap[m][k + 1]; }
            const float* Bp = Bp0 + (size_t)k * Ncols;
            v2f b[NT];
#pragma unroll
            for (int t = 0; t < NT; ++t) { b[t].x = Bp[t * 16]; b[t].y = Bp[t * 16 + Ncols]; }
            // ---- then the MT*NT WMMAs ----
#pragma unroll
            for (int m = 0; m < MT; ++m)
#pragma unroll
                for (int t = 0; t < NT; ++t)
                    acc[m][t] = __builtin_amdgcn_wmma_f32_16x16x4_f32(
                        false, a[m], false, b[t], (short)0, acc[m][t], false, false);
        }
    }
#pragma unroll
    for (int m = 0; m < MT; ++m) {
        const int rbase = row0 + m * 16;
        if (rbase < M) {                            // wave-uniform tail skip
#pragma unroll
            for (int t = 0; t < NT; ++t) {
                const int col = col0 + t * 16 + l16;
                const float bv = bias[col];
#pragma unroll
                for (int r = 0; r < 8; ++r) {
                    float val = acc[m][t][r] + bv;
                    if constexpr (RELU) val = fmaxf(val, 0.f);
                    C[(size_t)(rbase + r + half * 8) * ldc + col] = val;
                }
            }
        }
    }
}

// --------------------------- LayerNorm (D=256), one wave per row ------------
__global__ void layernorm_k(const float* __restrict__ in, float* __restrict__ out,
                            const float* __restrict__ g, const float* __restrict__ b,
                            int M) {
    const int wid  = (blockIdx.x * blockDim.x + threadIdx.x) >> 5;
    const int lane = threadIdx.x & 31;
    if (wid >= M) return;
    const float* row = in + (size_t)wid * D_MODEL;
    float v[8];
    float s = 0.f;
#pragma unroll
    for (int i = 0; i < 8; ++i) { v[i] = row[lane + 32 * i]; s += v[i]; }
    s = wave_sum32(s);
    const float mean = s * (1.f / 256.f);
    float vs = 0.f;
#pragma unroll
    for (int i = 0; i < 8; ++i) { const float d = v[i] - mean; vs += d * d; }
    vs = wave_sum32(vs);
    const float rstd = rsqrtf(vs * (1.f / 256.f) + 1e-5f);
    float* orow = out + (size_t)wid * D_MODEL;
#pragma unroll
    for (int i = 0; i < 8; ++i) {
        const int c = lane + 32 * i;
        orow[c] = (v[i] - mean) * rstd * g[c] + b[c];
    }
}

// ---- Fused action-net projections: s8[n] = h[n] @ [Wir|Wor|Wia|Woa] --------
// per node: {in_r0,in_r1,out_r0,out_r1,in_a0,in_a1,out_a0,out_a1}
__global__ void smallproj_k(const float* __restrict__ h,
                            const float* __restrict__ Wir, const float* __restrict__ Wia,
                            const float* __restrict__ Wor, const float* __restrict__ Woa,
                            float* __restrict__ s8, int M) {
    const int wid  = (blockIdx.x * blockDim.x + threadIdx.x) >> 5;
    const int lane = threadIdx.x & 31;
    if (wid >= M) return;
    const float* row = h + (size_t)wid * D_MODEL;
    float a[8] = {0.f, 0.f, 0.f, 0.f, 0.f, 0.f, 0.f, 0.f};
#pragma unroll
    for (int i = 0; i < 8; ++i) {
        const int c = lane + 32 * i;
        const float hv = row[c];
        a[0] += hv * Wir[2 * c];  a[1] += hv * Wir[2 * c + 1];
        a[2] += hv * Wor[2 * c];  a[3] += hv * Wor[2 * c + 1];
        a[4] += hv * Wia[2 * c];  a[5] += hv * Wia[2 * c + 1];
        a[6] += hv * Woa[2 * c];  a[7] += hv * Woa[2 * c + 1];
    }
#pragma unroll
    for (int j = 0; j < 8; ++j) a[j] = wave_sum32(a[j]);
    if (lane == 0) {
        float* o = s8 + (size_t)wid * 8;
#pragma unroll
        for (int j = 0; j < 8; ++j) o[j] = a[j];
    }
}

// ============================ CSR construction ==============================
__global__ void edge_count_k(const int* __restrict__ ei, int* __restrict__ deg, int E) {
    const int stride = gridDim.x * blockDim.x;
    for (int e = blockIdx.x * blockDim.x + threadIdx.x; e < E; e += stride)
        atomicAdd(&deg[ei[e]], 1);
}

// block-local inclusive scan (256 elems/block) + per-block totals
__global__ void scan1_k(const int* __restrict__ deg, int* __restrict__ incl,
                        int* __restrict__ blockSums, int M) {
    __shared__ int sh[256];
    const int tid = threadIdx.x;
    const int i = blockIdx.x * 256 + tid;
    sh[tid] = (i < M) ? deg[i] : 0;
    __syncthreads();
#pragma unroll
    for (int off = 1; off < 256; off <<= 1) {
        int t = (tid >= off) ? sh[tid - off] : 0;
        __syncthreads();
        sh[tid] += t;
        __syncthreads();
    }
    if (i < M) incl[i] = sh[tid];
    if (tid == 255) blockSums[blockIdx.x] = sh[255];
}

// serial exclusive scan over block totals (nb ~ 196, trivial)
__global__ void scan2_k(int* __restrict__ blockSums, int* __restrict__ blockOffs, int nb) {
    if (threadIdx.x == 0 && blockIdx.x == 0) {
        int run = 0;
        for (int i = 0; i < nb; ++i) { blockOffs[i] = run; run += blockSums[i]; }
    }
}

// offsets[i+1] = incl[i] + blockOffs[block]; offsets[0] = 0; cursor = offsets
__global__ void scan3_k(const int* __restrict__ incl, const int* __restrict__ blockOffs,
                        int* __restrict__ offsets, int* __restrict__ cursor, int M) {
    const int i = blockIdx.x * blockDim.x + threadIdx.x;
    if (i == 0) { offsets[0] = 0; cursor[0] = 0; }
    if (i < M) {
        const int o = incl[i] + blockOffs[i >> 8];
        offsets[i + 1] = o;
        if (i + 1 < M) cursor[i + 1] = o;
    }
}

__global__ void scatter_csr_k(const int* __restrict__ ei, int* __restrict__ cursor,
                              int* __restrict__ csr, int E) {
    const int stride = gridDim.x * blockDim.x;
    for (int e = blockIdx.x * blockDim.x + threadIdx.x; e < E; e += stride) {
        const int pos = atomicAdd(&cursor[ei[e]], 1);
        csr[pos] = e;
    }
}

// insertion-sort each node's edge list by edge id -> deterministic CSR
__global__ void sort_csr_k(const int* __restrict__ offsets, int* __restrict__ csr, int M) {
    const int n = blockIdx.x * blockDim.x + threadIdx.x;
    if (n >= M) return;
    const int s = offsets[n], t = offsets[n + 1];
    for (int i = s + 1; i < t; ++i) {
        const int key = csr[i];
        int j = i - 1;
        while (j >= s && csr[j] > key) { csr[j + 1] = csr[j]; --j; }
        csr[j + 1] = key;
    }
}

// ==== gates: gather 4 logit channels over in-edges, fuse hard-Gumbel argmax =
__global__ void gate_gather_k(const int* __restrict__ ei, const int* __restrict__ offsets,
                              const int* __restrict__ csr,
                              const float* __restrict__ s8,
                              const float* __restrict__ b_in, const float* __restrict__ b_out,
                              const float* __restrict__ gn,  // layer slice [2,N,2]
                              float* __restrict__ gates, int M, int E) {
    const int n = blockIdx.x * blockDim.x + threadIdx.x;
    if (n >= M) return;
    float a0 = 0.f, a1 = 0.f, a2 = 0.f, a3 = 0.f;
    const int s = offsets[n], t = offsets[n + 1];
    for (int j = s; j < t; ++j) {
        const int v = ei[E + csr[j]];
        const float* sv = s8 + (size_t)v * 8 + 4;
        a0 += sv[0]; a1 += sv[1]; a2 += sv[2]; a3 += sv[3];
    }
    const float* sn = s8 + (size_t)n * 8;
    const float il0 = sn[0] + a0 + b_in[0];
    const float il1 = sn[1] + a1 + b_in[1];
    const float ol0 = sn[2] + a2 + b_out[0];
    const float ol1 = sn[3] + a3 + b_out[1];
    const float gi0 = gn[(size_t)n * 2 + 0],       gi1 = gn[(size_t)n * 2 + 1];
    const float go0 = gn[(size_t)(M + n) * 2 + 0], go1 = gn[(size_t)(M + n) * 2 + 1];
    gates[(size_t)n * 2 + 0] = ((il0 + gi0) >= (il1 + gi1)) ? 1.f : 0.f;
    gates[(size_t)n * 2 + 1] = ((ol0 + go0) >= (ol1 + go1)) ? 1.f : 0.f;
}

// ==== dense aggregation as a gather: agg[u] = gate_in[u]*sum gate_out[v]*h[v]
// one wave per destination node; no atomics; deterministic (sorted CSR order)
__global__ void dense_gather_k(const int* __restrict__ ei, const int* __restrict__ offsets,
                               const int* __restrict__ csr,
                               const float* __restrict__ gates,
                               const float* __restrict__ h, float* __restrict__ agg,
                               int M, int E) {
    const int u    = (blockIdx.x * blockDim.x + threadIdx.x) >> 5;
    const int lane = threadIdx.x & 31;
    if (u >= M) return;
    float acc[8] = {0.f, 0.f, 0.f, 0.f, 0.f, 0.f, 0.f, 0.f};
    if (gates[(size_t)u * 2] != 0.f) {
        const int s = offsets[u], t = offsets[u + 1];
        for (int j = s; j < t; ++j) {                    // wave-uniform loop
            const int v = ei[E + csr[j]];
            if (gates[(size_t)v * 2 + 1] != 0.f) {
                const float* hv = h + (size_t)v * D_MODEL;
#pragma unroll
                for (int i = 0; i < 8; ++i) acc[i] += hv[lane + 32 * i];
            }
        }
    }
    float* au = agg + (size_t)u * D_MODEL;
#pragma unroll
    for (int i = 0; i < 8; ++i) au[lane + 32 * i] = acc[i];
}

// ---------------------------------------------------------------------------
extern "C" void kernel_launch(void* const* d_in, const int* in_sizes, int n_in,
                              void* d_out, int out_size, void* d_ws, size_t ws_size,
                              hipStream_t stream) {
    const float* x     = (const float*)d_in[0];
    const int*   ei    = (const int*)d_in[1];
    const float* gn    = (const float*)d_in[2];
    const float* W_enc = (const float*)d_in[3];
    const float* b_enc = (const float*)d_in[4];
    const float* W_root= (const float*)d_in[5];
    const float* W_agg = (const float*)d_in[6];
    const float* b_env = (const float*)d_in[7];
    const float* Wir   = (const float*)d_in[8];
    const float* Wia   = (const float*)d_in[9];
    const float* b_in  = (const float*)d_in[10];
    const float* Wor   = (const float*)d_in[11];
    const float* Woa   = (const float*)d_in[12];
    const float* b_out = (const float*)d_in[13];
    // d_in[14..16] (Wt_r, Wt_a, b_t): dead in the forward value (temp>0)
    const float* ln_g  = (const float*)d_in[17];
    const float* ln_b  = (const float*)d_in[18];
    const float* W_dec = (const float*)d_in[19];
    const float* b_dec = (const float*)d_in[20];

    const int N = in_sizes[0] / D_INP;                  // 50000 (multiple of 16)
    const int E = in_sizes[1] / 2;                      // 800000
    const int L = in_sizes[5] / (D_MODEL * D_MODEL);    // 3
    const int D_OUT = in_sizes[20];                     // 64
    const int nb = (N + 255) / 256;                     // scan blocks

    // ---- workspace carve-up ----
    float* hA    = (float*)d_ws;
    float* hB    = hA  + (size_t)N * D_MODEL;
    float* agg   = hB  + (size_t)N * D_MODEL;
    float* s8    = agg + (size_t)N * D_MODEL;
    float* gates = s8  + (size_t)N * 8;
    int* offsets   = (int*)(gates + (size_t)N * 2);
    int* cursor    = offsets + (N + 1);     // doubles as degree buffer
    int* incl      = cursor + N;
    int* blockSums = incl + N;
    int* blockOffs = blockSums + nb;
    int* csr       = blockOffs + nb;

    const int TB = 256;
    const int wpb = TB / 32;                               // 8 waves/block
    const int waveBlocks = (N + wpb - 1) / wpb;
    const int nodeBlocks = (N + TB - 1) / TB;

    // ---- CSR build (deterministic: per-node lists sorted by edge id) ----
    hipMemsetAsync(cursor, 0, (size_t)N * sizeof(int), stream);
    edge_count_k<<<1024, TB, 0, stream>>>(ei, cursor, E);
    scan1_k<<<nb, 256, 0, stream>>>(cursor, incl, blockSums, N);
    scan2_k<<<1, 64, 0, stream>>>(blockSums, blockOffs, nb);
    scan3_k<<<nodeBlocks + 1, TB, 0, stream>>>(incl, blockOffs, offsets, cursor, N);
    scatter_csr_k<<<1024, TB, 0, stream>>>(ei, cursor, csr, E);
    sort_csr_k<<<nodeBlocks, TB, 0, stream>>>(offsets, csr, N);

    // ---- encoder: hA = relu(x @ W_enc + b_enc) ----
    {
        const int groups = ((N / 16 + 1) / 2) * (D_MODEL / 64);
        gemm_wmma_f32<2, 4, false, true><<<(groups + wpb - 1) / wpb, TB, 0, stream>>>(
            x, D_INP, W_enc, nullptr, nullptr, b_enc, hA, D_MODEL, N, D_INP, D_MODEL);
    }

    for (int l = 0; l < L; ++l) {
        layernorm_k<<<waveBlocks, TB, 0, stream>>>(hA, hA, ln_g, ln_b, N);
        smallproj_k<<<waveBlocks, TB, 0, stream>>>(hA, Wir, Wia, Wor, Woa, s8, N);
        gate_gather_k<<<nodeBlocks, TB, 0, stream>>>(ei, offsets, csr, s8, b_in, b_out,
                                                     gn + (size_t)l * 2 * N * 2, gates, N, E);
        dense_gather_k<<<waveBlocks, TB, 0, stream>>>(ei, offsets, csr, gates, hA, agg, N, E);

        const int groups = ((N / 16 + 1) / 2) * (D_MODEL / 64);
        gemm_wmma_f32<2, 4, true, true><<<(groups + wpb - 1) / wpb, TB, 0, stream>>>(
            hA, D_MODEL, W_root + (size_t)l * D_MODEL * D_MODEL,
            agg, W_agg + (size_t)l * D_MODEL * D_MODEL,
            b_env + (size_t)l * D_MODEL, hB, D_MODEL, N, D_MODEL, D_MODEL);

        float* t = hA; hA = hB; hB = t;   // host-side ping-pong (deterministic)
    }

    layernorm_k<<<waveBlocks, TB, 0, stream>>>(hA, hA, ln_g, ln_b, N);
    {
        const int groups = ((N / 16 + 1) / 2) * (D_OUT / 64);
        gemm_wmma_f32<2, 4, false, false><<<(groups + wpb - 1) / wpb, TB, 0, stream>>>(
            hA, D_MODEL, W_dec, nullptr, nullptr, b_dec,
            (float*)d_out, D_OUT, N, D_MODEL, D_OUT);
    }
}